// FusedSinkhornKnopp_25056839205186
// MI455X (gfx1250) — compile-verified
//
#include <hip/hip_runtime.h>
#include <cstdint>

#define B_MATS 16
#define N      2048
#define ITERS  20
#define EPS    1e-12f

typedef __attribute__((ext_vector_type(2))) float v2f;
typedef __attribute__((ext_vector_type(8))) float v8f;

// ---- CDNA5 async global->LDS staging (ASYNCcnt path) --------------------
__device__ __forceinline__ void async_lds_b128(const float4* src, float4* dst) {
    unsigned lds           = (unsigned)(uintptr_t)dst;     // flat LDS addr: low 32 bits = LDS offset
    unsigned long long gad = (unsigned long long)(uintptr_t)src;
    asm volatile("global_load_async_to_lds_b128 %0, %1, off"
                 :: "v"(lds), "v"(gad) : "memory");
}
__device__ __forceinline__ void async_wait0() {
    asm volatile("s_wait_asynccnt 0" ::: "memory");
}

// ---- E = exp(x) ----------------------------------------------------------
__global__ void initE(const float* __restrict__ in, float* __restrict__ out,
                      int matBase, int matCount) {
    const size_t base4 = (size_t)matBase * N * N / 4;
    const size_t n4    = (size_t)matCount * N * N / 4;
    const float4* in4  = (const float4*)in;
    float4*       out4 = (float4*)out;
    for (size_t i = (size_t)blockIdx.x * blockDim.x + threadIdx.x; i < n4;
         i += (size_t)gridDim.x * blockDim.x) {
        float4 v = in4[base4 + i];
        v.x = expf(v.x); v.y = expf(v.y); v.z = expf(v.z); v.w = expf(v.w);
        out4[base4 + i] = v;
    }
}

__global__ void initVec(float* __restrict__ p, int n) {
    int i = blockIdx.x * blockDim.x + threadIdx.x;
    if (i < n) p[i] = 1.0f;
}

// ---- row phase: r' = r / (r * (E c) + eps) ------------------------------
// 256 threads = 8 waves; wave handles 4 rows; 64 blocks per matrix.
__global__ void __launch_bounds__(256)
rowPhase(const float* __restrict__ E, float* __restrict__ r,
         const float* __restrict__ c, int matBase) {
    __shared__ float cs[N];
    const int mat      = matBase + (int)(blockIdx.x >> 6);
    const int rowBlock = ((int)blockIdx.x & 63) * 32;
    const int t = threadIdx.x;

    {   // stage c[mat] (8 KB) into LDS with async copies
        const float4* csrc = (const float4*)(c + (size_t)mat * N);
        float4*       cdst = (float4*)cs;
        async_lds_b128(csrc + t,       cdst + t);
        async_lds_b128(csrc + t + 256, cdst + t + 256);
        async_wait0();
    }
    __syncthreads();

    const int lane = t & 31, wave = t >> 5;
    const float4* cp = (const float4*)cs;
    for (int rr = 0; rr < 4; ++rr) {
        const int row = rowBlock + wave * 4 + rr;
        const float4* rp = (const float4*)(E + (size_t)mat * N * N + (size_t)row * N);
        float s = 0.f;
#pragma unroll
        for (int k = 0; k < N / 128; ++k) {
            float4 a = rp[lane + 32 * k];
            float4 b = cp[lane + 32 * k];
            s += a.x * b.x + a.y * b.y + a.z * b.z + a.w * b.w;
        }
        s += __shfl_down(s, 16);
        s += __shfl_down(s, 8);
        s += __shfl_down(s, 4);
        s += __shfl_down(s, 2);
        s += __shfl_down(s, 1);
        if (lane == 0) {
            float* rv = r + (size_t)mat * N + row;
            float old = *rv;
            *rv = old / (old * s + EPS);
        }
    }
}

// ---- col phase (WMMA f32): c' = c / (c * (E^T r) + eps) -----------------
// 128 threads = 4 waves; each wave owns 16 columns; 32 blocks per matrix.
// A(16x4) = broadcast r[i0..i0+3]  -> every row of D = partial colsums.
// B(4x16) = E[i0..i0+3, j0..j0+15] per the ISA B-matrix lane layout.
__global__ void __launch_bounds__(128)
colPhase(const float* __restrict__ E, const float* __restrict__ r,
         float* __restrict__ c, int matBase) {
    __shared__ float rs[N];
    const int mat      = matBase + (int)(blockIdx.x >> 5);
    const int colBlock = ((int)blockIdx.x & 31) * 64;
    const int t = threadIdx.x;

    {   // stage r[mat] (8 KB) into LDS with async copies
        const float4* rsrc = (const float4*)(r + (size_t)mat * N);
        float4*       rdst = (float4*)rs;
#pragma unroll
        for (int k = 0; k < 4; ++k)
            async_lds_b128(rsrc + t + 128 * k, rdst + t + 128 * k);
        async_wait0();
    }
    __syncthreads();

    const int lane = t & 31, wave = t >> 5;
    const int l16  = lane & 15;
    const int hv   = lane >> 4;                 // 0: K=0,1   1: K=2,3
    const int j0   = colBlock + wave * 16;
    const float*  Ep  = E + (size_t)mat * N * N + j0 + l16;
    const float2* rp2 = (const float2*)rs;

    v8f acc = {0.f, 0.f, 0.f, 0.f, 0.f, 0.f, 0.f, 0.f};
#pragma unroll 4
    for (int i0 = 0; i0 < N; i0 += 4) {
        float2 av = rp2[(i0 >> 1) + hv];        // {r[i0+2hv], r[i0+2hv+1]}
        v2f a; a.x = av.x; a.y = av.y;
        const float* bp = Ep + (size_t)(i0 + 2 * hv) * N;
        v2f b; b.x = bp[0]; b.y = bp[N];
        acc = __builtin_amdgcn_wmma_f32_16x16x4_f32(
            false, a, false, b, (short)0, acc, false, false);
    }
    if (lane < 16) {                            // D row 0 = colsum for j0+l16
        float S = acc[0];
        float* cp = c + (size_t)mat * N + j0 + l16;
        float cv = *cp;
        *cp = cv / (cv * S + EPS);
    }
}

// ---- finalize: out = r .* E .* c (in place over E stored in d_out) ------
__global__ void finalizeK(float* __restrict__ out, const float* __restrict__ r,
                          const float* __restrict__ c, int matBase, int matCount) {
    const size_t base4 = (size_t)matBase * N * N / 4;
    const size_t n4    = (size_t)matCount * N * N / 4;
    const size_t NN4   = (size_t)N * N / 4;
    float4*       out4 = (float4*)out;
    const float4* c4   = (const float4*)c;
    for (size_t i = (size_t)blockIdx.x * blockDim.x + threadIdx.x; i < n4;
         i += (size_t)gridDim.x * blockDim.x) {
        size_t g   = base4 + i;
        int    mat = (int)(g / NN4);
        size_t rem = g - (size_t)mat * NN4;
        int    row = (int)(rem / (N / 4));
        int    c4i = (int)(rem % (N / 4));
        float  rv  = r[(size_t)mat * N + row];
        float4 cv  = c4[(size_t)mat * (N / 4) + c4i];
        float4 v   = out4[g];
        v.x *= rv * cv.x; v.y *= rv * cv.y; v.z *= rv * cv.z; v.w *= rv * cv.w;
        out4[g] = v;
    }
}

extern "C" void kernel_launch(void* const* d_in, const int* in_sizes, int n_in,
                              void* d_out, int out_size, void* d_ws, size_t ws_size,
                              hipStream_t stream) {
    (void)in_sizes; (void)n_in; (void)out_size; (void)ws_size;
    const float* in  = (const float*)d_in[0];
    float*       out = (float*)d_out;
    float*       r   = (float*)d_ws;               // 16*2048 f32
    float*       c   = r + (size_t)B_MATS * N;     // 16*2048 f32

    initVec<<<(2 * B_MATS * N) / 256, 256, 0, stream>>>(r, 2 * B_MATS * N);

    // Two halves of 8 matrices (128 MB) so all 40 sweeps of a half stay
    // resident in the 192 MB L2.
    for (int h = 0; h < 2; ++h) {
        const int mb = h * 8, mc = 8;
        initE<<<4096, 256, 0, stream>>>(in, out, mb, mc);
        for (int it = 0; it < ITERS; ++it) {
            rowPhase<<<mc * 64, 256, 0, stream>>>(out, r, c, mb);
            colPhase<<<mc * 32, 128, 0, stream>>>(out, r, c, mb);
        }
        finalizeK<<<4096, 256, 0, stream>>>(out, r, c, mb, mc);
    }
}